// MeshGNN_45268955299958
// MI455X (gfx1250) — compile-verified
//
#include <hip/hip_runtime.h>

typedef __attribute__((ext_vector_type(2))) float v2f;
typedef __attribute__((ext_vector_type(8))) float v8f;

#define D 128
#define WS_STRIDE 136   // pad LDS row stride: 2 rows apart -> 272 % 64 = 16 banks offset

// ---------------- degree / norm ----------------

__global__ void k_deg_init(unsigned* __restrict__ deg, int n) {
    int i = blockIdx.x * blockDim.x + threadIdx.x;
    if (i < n) deg[i] = 1u;                      // self-loop contributes 1
}

__global__ void k_deg_count(const int* __restrict__ dst, unsigned* __restrict__ deg, int e) {
    int i = blockIdx.x * blockDim.x + threadIdx.x;
    if (i < e) atomicAdd(&deg[dst[i]], 1u);
}

__global__ void k_dinv(const unsigned* __restrict__ deg, float* __restrict__ dinv, int n) {
    int i = blockIdx.x * blockDim.x + threadIdx.x;
    if (i < n) dinv[i] = __frsqrt_rn((float)deg[i]);   // deg >= 1 always (self loops)
}

// ---------------- GEMM: H = X @ W  (fp32 WMMA) ----------------
// block = 256 threads = 8 wave32; wave w owns column tile n0 = 16*w,
// block owns row tile row0 = 16*blockIdx.x. K fully unrolled: 32 WMMAs/wave.
__global__ void __launch_bounds__(256) k_gemm_wmma(const float* __restrict__ X,
                                                   const float* __restrict__ W,
                                                   float* __restrict__ H) {
    __shared__ float Ws[D * WS_STRIDE];

    const int tid = threadIdx.x;
    // stage W (128x128 f32) into LDS, float4 per transaction, padded stride
    const float4* Wv = (const float4*)W;
#pragma unroll
    for (int i = 0; i < 16; ++i) {
        int idx4 = tid + 256 * i;                // 0..4095 float4's
        float4 v = Wv[idx4];
        int row = idx4 >> 5;                     // (idx4*4)/128
        int col = (idx4 & 31) << 2;
        *(float4*)&Ws[row * WS_STRIDE + col] = v;
    }
    __syncthreads();

    const int lane = tid & 31;
    const int wave = tid >> 5;
    const int n0   = wave << 4;                  // column tile base
    const int m    = lane & 15;
    const int kk   = (lane >> 4) << 1;           // 0 for lanes 0-15, 2 for lanes 16-31
    const int row0 = blockIdx.x << 4;

    const float* xrow = X + (size_t)(row0 + m) * D;

    v8f c = {0.f, 0.f, 0.f, 0.f, 0.f, 0.f, 0.f, 0.f};
#pragma unroll
    for (int k = 0; k < D; k += 4) {
        // A frag: lane m holds X[row0+m][k+kk .. k+kk+1]
        v2f a = *(const v2f*)(xrow + k + kk);
        // B frag: lane m holds W[k+kk][n0+m], W[k+kk+1][n0+m]
        v2f b;
        b.x = Ws[(k + kk)     * WS_STRIDE + n0 + m];
        b.y = Ws[(k + kk + 1) * WS_STRIDE + n0 + m];
        c = __builtin_amdgcn_wmma_f32_16x16x4_f32(false, a, false, b,
                                                  (short)0, c, false, false);
    }

    // C/D layout: vgpr r -> row (row0 + r + 8*(lane>>4)), col (n0 + (lane&15))
    float* out = H + (size_t)(row0 + (lane >> 4) * 8) * D + n0 + m;
#pragma unroll
    for (int r = 0; r < 8; ++r)
        out[(size_t)r * D] = c[r];
}

// ---------------- self-loop + bias epilogue ----------------
// out[i][d] = h[i][d] * dinv[i]^2 + b[d]   (fully overwrites poisoned output)
__global__ void k_selfloop_bias(const float* __restrict__ h,
                                const float* __restrict__ dinv,
                                const float* __restrict__ bias,
                                float* __restrict__ out, int n) {
    long long idx = (long long)blockIdx.x * blockDim.x + threadIdx.x;
    if (idx < (long long)n * D) {
        int i = (int)(idx >> 7);
        int d = (int)(idx & (D - 1));
        float s = dinv[i];
        out[idx] = h[idx] * (s * s) + bias[d];
    }
}

// ---------------- edge scatter: out[dst] += h[src] * dinv[src]*dinv[dst] ----------------
// one wave32 per edge; each lane owns one float4 (32 * 16B = 512B row)
__global__ void __launch_bounds__(256) k_scatter(const int* __restrict__ src,
                                                 const int* __restrict__ dst,
                                                 const float* __restrict__ dinv,
                                                 const float* __restrict__ h,
                                                 float* __restrict__ out, int e) {
    int gid  = blockIdx.x * blockDim.x + threadIdx.x;
    int edge = gid >> 5;
    int lane = threadIdx.x & 31;
    if (edge >= e) return;                        // wave-uniform exit

    int s = src[edge];
    int d = dst[edge];
    float norm = dinv[s] * dinv[d];

    float4 v = ((const float4*)(h + (size_t)s * D))[lane];   // L2-resident gather
    float* o = out + (size_t)d * D + (lane << 2);
    unsafeAtomicAdd(o + 0, v.x * norm);           // non-returning global_atomic_add_f32
    unsafeAtomicAdd(o + 1, v.y * norm);
    unsafeAtomicAdd(o + 2, v.z * norm);
    unsafeAtomicAdd(o + 3, v.w * norm);
}

// ---------------- launcher ----------------
extern "C" void kernel_launch(void* const* d_in, const int* in_sizes, int n_in,
                              void* d_out, int out_size, void* d_ws, size_t ws_size,
                              hipStream_t stream) {
    const float* x  = (const float*)d_in[0];
    const int*   ei = (const int*)d_in[1];
    const float* W1 = (const float*)d_in[2];
    const float* b1 = (const float*)d_in[3];
    const float* W2 = (const float*)d_in[4];
    const float* b2 = (const float*)d_in[5];
    float* out = (float*)d_out;

    const int n = in_sizes[0] / D;       // 100000
    const int e = in_sizes[1] / 2;       // 1600000
    const int* src = ei;                 // edge_index[0]
    const int* dst = ei + e;             // edge_index[1]

    // workspace layout (all 16B-aligned for n multiple of 4)
    float*    ws   = (float*)d_ws;
    unsigned* deg  = (unsigned*)ws;                      // n u32
    float*    dinv = ws + n;                             // n f32
    float*    h    = ws + 2 * (size_t)n;                 // n*128 f32
    float*    out1 = h + (size_t)n * D;                  // n*128 f32

    const int T = 256;
    dim3 blk(T);

    // normalization coefficients
    k_deg_init <<<(n + T - 1) / T, blk, 0, stream>>>(deg, n);
    k_deg_count<<<(e + T - 1) / T, blk, 0, stream>>>(dst, deg, e);
    k_dinv     <<<(n + T - 1) / T, blk, 0, stream>>>(deg, dinv, n);

    const int gemm_blocks = n / 16;                      // 6250 (exact)
    const int elem_blocks = (int)(((long long)n * D + T - 1) / T);
    const int edge_blocks = (e * 32 + T - 1) / T;        // wave per edge

    // layer 1
    k_gemm_wmma   <<<gemm_blocks, blk, 0, stream>>>(x, W1, h);
    k_selfloop_bias<<<elem_blocks, blk, 0, stream>>>(h, dinv, b1, out1, n);
    k_scatter     <<<edge_blocks, blk, 0, stream>>>(src, dst, dinv, h, out1, e);

    // layer 2
    k_gemm_wmma   <<<gemm_blocks, blk, 0, stream>>>(out1, W2, h);
    k_selfloop_bias<<<elem_blocks, blk, 0, stream>>>(h, dinv, b2, out, n);
    k_scatter     <<<edge_blocks, blk, 0, stream>>>(src, dst, dinv, h, out, e);
}